// GraphSAGEIDS_55980603736252
// MI455X (gfx1250) — compile-verified
//
#include <hip/hip_runtime.h>
#include <hip/hip_bf16.h>

typedef __attribute__((ext_vector_type(2))) float v2f;
typedef __attribute__((ext_vector_type(8))) float v8f;

#define EPS 1e-5f

// ---------------------------------------------------------------- utilities

__global__ void k_zero(float* __restrict__ p, long n) {
    long i = (long)blockIdx.x * blockDim.x + threadIdx.x;
    if (i < n) p[i] = 0.0f;
}

__global__ void k_deg_count(const int* __restrict__ ei, float* __restrict__ deg, int E) {
    int e = blockIdx.x * blockDim.x + threadIdx.x;
    if (e < E) atomicAdd(&deg[ei[(long)E + e]], 1.0f);
}

// deg[n] <- 1 / max(deg[n], 1)
__global__ void k_deg_recip(float* __restrict__ deg, int n) {
    int i = blockIdx.x * blockDim.x + threadIdx.x;
    if (i < n) deg[i] = 1.0f / fmaxf(deg[i], 1.0f);
}

// --------------------------------------------------- edge scatter-add (mean numerator)
// One thread per (edge, 4-channel chunk). For D=128 a wave covers exactly one
// edge (128B*4 coalesced loads); atomics go to agg[dst].

template <int D>
__global__ void k_scatter_add(const float* __restrict__ x, const int* __restrict__ ei,
                              float* __restrict__ agg, int E) {
    constexpr int Q = D / 4;
    long t = (long)blockIdx.x * blockDim.x + threadIdx.x;
    if (t >= (long)E * Q) return;
    int e = (int)(t / Q);
    int q = (int)(t - (long)e * Q);
    int src = ei[e];
    int dst = ei[(long)E + e];
    const float4 v = *(const float4*)(x + (long)src * D + 4 * q);
    float* a = agg + (long)dst * D + 4 * q;
    atomicAdd(a + 0, v.x);
    atomicAdd(a + 1, v.y);
    atomicAdd(a + 2, v.z);
    atomicAdd(a + 3, v.w);
}

// ------------------------------------------------------------- SAGE GEMM (WMMA fp32)
// pre[n,o] = sum_k (agg[n,k]*rinv[n]) * wl[o,k] + sum_k h[n,k] * wr[o,k] + bl[o]
// Block = DOUT/16 waves; wave w owns output columns [16w,16w+16); block owns a
// 16-node row tile. Fragments follow the CDNA5 ISA layouts:
//   A 16x4 f32: lane l -> row l&15, K = k0 + 2*(l>>4) + {reg0,reg1}
//   B 4x16 f32: lane l -> col l&15, K = k0 + 2*(l>>4) + {reg0,reg1}
//   C/D:        reg r  -> row r + 8*(l>>4), col l&15

template <int DIN, int DOUT>
__global__ __launch_bounds__(DOUT * 2) void k_sage_gemm(
    const float* __restrict__ agg, const float* __restrict__ rinv,
    const float* __restrict__ h, const float* __restrict__ wl,
    const float* __restrict__ bl, const float* __restrict__ wr,
    float* __restrict__ pre, int n) {
    const int lane = threadIdx.x & 31;
    const int wv = threadIdx.x >> 5;     // wave id -> column tile
    const int obase = wv * 16;
    const int half = lane >> 4;          // 0 / 1 (K-pair select)
    const int lid = lane & 15;           // row (A) / col (B,C)
    const int oc = obase + lid;

    const long row = (long)blockIdx.x * 16 + lid;
    const long rowc = (row < n) ? row : (long)n - 1;   // clamp (keeps EXEC full)
    const float r = rinv[rowc];

    const float bias = bl[oc];
    v8f acc;
#pragma unroll
    for (int i = 0; i < 8; ++i) acc[i] = bias;

    const float* __restrict__ aRow = agg + rowc * DIN;
    const float* __restrict__ hRow = h + rowc * DIN;
    const float* __restrict__ wlRow = wl + (long)oc * DIN;
    const float* __restrict__ wrRow = wr + (long)oc * DIN;

#pragma unroll 4
    for (int k0 = 0; k0 < DIN; k0 += 4) {
        const int k = k0 + 2 * half;
        v2f a, b;
        // mean-path: scale numerator by 1/deg on the fly
        a[0] = aRow[k] * r;  a[1] = aRow[k + 1] * r;
        b[0] = wlRow[k];     b[1] = wlRow[k + 1];
        acc = __builtin_amdgcn_wmma_f32_16x16x4_f32(false, a, false, b, (short)0, acc,
                                                    false, false);
        // root-path
        a[0] = hRow[k];  a[1] = hRow[k + 1];
        b[0] = wrRow[k]; b[1] = wrRow[k + 1];
        acc = __builtin_amdgcn_wmma_f32_16x16x4_f32(false, a, false, b, (short)0, acc,
                                                    false, false);
    }

    float* __restrict__ outBase = pre + (long)blockIdx.x * 16 * DOUT + oc;
#pragma unroll
    for (int i = 0; i < 8; ++i) {
        const int m = i + 8 * half;
        if ((long)blockIdx.x * 16 + m < n) outBase[(long)m * DOUT] = acc[i];
    }
}

// ------------------------------------------------------------------ batch norm
// Pass 1: per-channel sum / sumsq (thread = channel, coalesced rows, block partials)
__global__ void k_bn_stats(const float* __restrict__ x, float* __restrict__ stats,
                           int n, int C) {
    const int c = threadIdx.x;   // blockDim.x == C
    float s = 0.0f, ss = 0.0f;
    for (long row = blockIdx.x; row < n; row += gridDim.x) {
        const float v = x[row * C + c];
        s += v;
        ss += v * v;
    }
    atomicAdd(&stats[c], s);
    atomicAdd(&stats[C + c], ss);
}

// Pass 2: fold mean/var/gamma/beta into per-channel scale+shift (in place)
__global__ void k_bn_finalize(float* __restrict__ stats, const float* __restrict__ g,
                              const float* __restrict__ be, int n, int C) {
    const int c = blockIdx.x * blockDim.x + threadIdx.x;
    if (c >= C) return;
    const float inv_n = 1.0f / (float)n;
    const float mu = stats[c] * inv_n;
    const float var = stats[C + c] * inv_n - mu * mu;   // biased batch variance
    const float sc = g[c] * rsqrtf(var + EPS);
    const float sh = be[c] - mu * sc;
    stats[c] = sc;
    stats[C + c] = sh;
}

// Pass 3: y = relu(x*scale + shift), in place. C is a power of two (128/256).
__global__ void k_bn_apply(float* __restrict__ x, const float* __restrict__ stats,
                           long total, int C) {
    long i = (long)blockIdx.x * blockDim.x + threadIdx.x;
    if (i >= total) return;
    const int c = (int)(i & (long)(C - 1));
    x[i] = fmaxf(fmaf(x[i], stats[c], stats[C + c]), 0.0f);
}

// ------------------------------------------------------------------ classifier
// out[n, 0..14] = h[n, :128] @ wc^T + bc, via WMMA with one padded zero column.
__global__ __launch_bounds__(256) void k_classifier(const float* __restrict__ h,
                                                    const float* __restrict__ wc,
                                                    const float* __restrict__ bc,
                                                    float* __restrict__ out,
                                                    int n, int ntiles) {
    const int lane = threadIdx.x & 31;
    const int wv = threadIdx.x >> 5;
    const int tile = blockIdx.x * 8 + wv;
    if (tile >= ntiles) return;   // wave-uniform; surviving waves keep EXEC full

    const int half = lane >> 4;
    const int lid = lane & 15;
    const bool valid_o = lid < 15;

    const long row = (long)tile * 16 + lid;
    const long rowc = (row < n) ? row : (long)n - 1;

    v8f acc;
    const float bias = valid_o ? bc[lid] : 0.0f;
#pragma unroll
    for (int i = 0; i < 8; ++i) acc[i] = bias;

    const float* __restrict__ hRow = h + rowc * 128;
    const float* __restrict__ wRow = wc + (long)(valid_o ? lid : 0) * 128;

#pragma unroll 4
    for (int k0 = 0; k0 < 128; k0 += 4) {
        const int k = k0 + 2 * half;
        v2f a, b;
        a[0] = hRow[k];
        a[1] = hRow[k + 1];
        const float b0 = wRow[k], b1 = wRow[k + 1];
        b[0] = valid_o ? b0 : 0.0f;   // pad column 15 with zeros (select, no branch)
        b[1] = valid_o ? b1 : 0.0f;
        acc = __builtin_amdgcn_wmma_f32_16x16x4_f32(false, a, false, b, (short)0, acc,
                                                    false, false);
    }

    if (valid_o) {
#pragma unroll
        for (int i = 0; i < 8; ++i) {
            const long m = (long)tile * 16 + i + 8 * half;
            if (m < n) out[m * 15 + lid] = acc[i];
        }
    }
}

// ------------------------------------------------------------------ host driver

static inline int cdiv(long a, int b) { return (int)((a + b - 1) / b); }

extern "C" void kernel_launch(void* const* d_in, const int* in_sizes, int n_in,
                              void* d_out, int out_size, void* d_ws, size_t ws_size,
                              hipStream_t stream) {
    const float* x    = (const float*)d_in[0];
    const int*   ei   = (const int*)d_in[1];
    const float* w1l  = (const float*)d_in[2];
    const float* b1l  = (const float*)d_in[3];
    const float* w1r  = (const float*)d_in[4];
    const float* g1   = (const float*)d_in[5];
    const float* be1  = (const float*)d_in[6];
    const float* w2l  = (const float*)d_in[7];
    const float* b2l  = (const float*)d_in[8];
    const float* w2r  = (const float*)d_in[9];
    const float* g2   = (const float*)d_in[10];
    const float* be2  = (const float*)d_in[11];
    const float* w3l  = (const float*)d_in[12];
    const float* b3l  = (const float*)d_in[13];
    const float* w3r  = (const float*)d_in[14];
    const float* g3   = (const float*)d_in[15];
    const float* be3  = (const float*)d_in[16];
    const float* wcl  = (const float*)d_in[17];
    const float* bcl  = (const float*)d_in[18];
    float* out = (float*)d_out;

    const int N = in_sizes[0] / 128;
    const int E = in_sizes[1] / 2;
    const int ntiles = (N + 15) / 16;

    float* slabA = (float*)d_ws;                 // agg (N x 256)
    float* slabB = slabA + (size_t)N * 256;      // rotating feature slab
    float* slabC = slabB + (size_t)N * 256;      // rotating feature slab
    float* deg   = slabC + (size_t)N * 256;      // N (becomes 1/max(deg,1))
    float* stats = deg + N;                      // 2*256

    const int T = 256;

    // degrees -> reciprocal (shared across all 3 layers)
    k_zero<<<cdiv(N, T), T, 0, stream>>>(deg, N);
    k_deg_count<<<cdiv(E, T), T, 0, stream>>>(ei, deg, E);
    k_deg_recip<<<cdiv(N, T), T, 0, stream>>>(deg, N);

    // ---------------- layer 1: x(128) -> slabB(256)
    k_zero<<<cdiv((long)N * 128, T), T, 0, stream>>>(slabA, (long)N * 128);
    k_scatter_add<128><<<cdiv((long)E * 32, T), T, 0, stream>>>(x, ei, slabA, E);
    k_sage_gemm<128, 256><<<ntiles, 512, 0, stream>>>(slabA, deg, x, w1l, b1l, w1r,
                                                      slabB, N);
    k_zero<<<cdiv(512, T), T, 0, stream>>>(stats, 512);
    k_bn_stats<<<512, 256, 0, stream>>>(slabB, stats, N, 256);
    k_bn_finalize<<<1, 256, 0, stream>>>(stats, g1, be1, N, 256);
    k_bn_apply<<<cdiv((long)N * 256, T), T, 0, stream>>>(slabB, stats, (long)N * 256, 256);

    // ---------------- layer 2: slabB(256) -> slabC(256)
    k_zero<<<cdiv((long)N * 256, T), T, 0, stream>>>(slabA, (long)N * 256);
    k_scatter_add<256><<<cdiv((long)E * 64, T), T, 0, stream>>>(slabB, ei, slabA, E);
    k_sage_gemm<256, 256><<<ntiles, 512, 0, stream>>>(slabA, deg, slabB, w2l, b2l, w2r,
                                                      slabC, N);
    k_zero<<<cdiv(512, T), T, 0, stream>>>(stats, 512);
    k_bn_stats<<<512, 256, 0, stream>>>(slabC, stats, N, 256);
    k_bn_finalize<<<1, 256, 0, stream>>>(stats, g2, be2, N, 256);
    k_bn_apply<<<cdiv((long)N * 256, T), T, 0, stream>>>(slabC, stats, (long)N * 256, 256);

    // ---------------- layer 3: slabC(256) -> slabB(128)
    k_zero<<<cdiv((long)N * 256, T), T, 0, stream>>>(slabA, (long)N * 256);
    k_scatter_add<256><<<cdiv((long)E * 64, T), T, 0, stream>>>(slabC, ei, slabA, E);
    k_sage_gemm<256, 128><<<ntiles, 256, 0, stream>>>(slabA, deg, slabC, w3l, b3l, w3r,
                                                      slabB, N);
    k_zero<<<cdiv(256, T), T, 0, stream>>>(stats, 256);
    k_bn_stats<<<512, 128, 0, stream>>>(slabB, stats, N, 128);
    k_bn_finalize<<<1, 128, 0, stream>>>(stats, g3, be3, N, 128);
    k_bn_apply<<<cdiv((long)N * 128, T), T, 0, stream>>>(slabB, stats, (long)N * 128, 128);

    // ---------------- classifier: slabB(128) -> out(15)
    k_classifier<<<cdiv(ntiles, 8), 256, 0, stream>>>(slabB, wcl, bcl, out, N, ntiles);
}